// LocalContrastNormalization_37194416783536
// MI455X (gfx1250) — compile-verified
//
#include <hip/hip_runtime.h>
#include <hip/hip_bf16.h>

typedef __attribute__((ext_vector_type(2))) float v2f;
typedef __attribute__((ext_vector_type(8))) float v8f;

#define B_  32
#define H_  512
#define W_  512
#define KS  9
#define RAD 4
#define TILE 16
#define PATCH 24          // TILE + KS - 1
#define PSTRIDE 26        // padded LDS row stride (even -> aligned b64, conflict-friendly)
#define WPAD 40           // padded weight-row stride, index shift +15
#define NCHUNK 54         // 9 dy * 6 K-chunks of 4
#define NPIX (B_ * H_ * W_)                    // 8,388,608 pixels
#define TILES_X (W_ / TILE)                    // 32
#define TILES_PER_IMG (TILES_X * (H_ / TILE))  // 1024
#define NTILES (B_ * TILES_PER_IMG)            // 32768
#define TILES_PER_BLK 8
#define NBLK (NTILES / TILES_PER_BLK)          // 4096

// Build lane-private banded-weight operands for all 54 WMMA K-chunks, once per
// block. bpack[c*32+lane] = the two B-matrix values lane contributes in chunk c.
__device__ __forceinline__ void build_bpack(const float* __restrict__ w,
                                            v2f* bpack, float* wl2, int lane)
{
    // zero-padded weight table: wl2[dy*40 + 15 + d] = w[dy][d] for d in [0,9)
    for (int i = lane; i < KS * WPAD; i += 32) {
        int dy = i / WPAD;
        int d  = i - dy * WPAD - 15;
        int dc = min(max(d, 0), KS - 1);
        float val = w[dy * KS + dc];              // clamped addr, unconditional load
        wl2[i] = (d >= 0 && d < KS) ? val : 0.0f;
    }
    __syncthreads();

    const int n    = lane & 15;
    const int base = (lane >> 4) << 1;            // K lane-group offset: 0 or 2
    #pragma unroll
    for (int c = 0; c < NCHUNK; ++c) {
        int dy = c / 6;
        int jc = (c - dy * 6) * 4;
        int d0 = jc + base - n + 15;              // in [0, 37] -> always in-range
        v2f bb;
        bb.x = wl2[dy * WPAD + d0];
        bb.y = wl2[dy * WPAD + d0 + 1];
        bpack[c * 32 + lane] = bb;
    }
    __syncthreads();
}

// One wave computes one 16x16 SAME-padded 9x9 conv tile via 54x
// V_WMMA_F32_16X16X4_F32. Inner loop: ds_load_b64(A) + ds_load_b64(B) + wmma.
__device__ __forceinline__ v8f wmma_conv_tile(const float* __restrict__ img,
                                              const v2f* bpack, float* patch,
                                              int y0, int x0, int lane)
{
    const bool interior = (y0 >= RAD) && (y0 + TILE + RAD <= H_) &&
                          (x0 >= RAD) && (x0 + TILE + RAD <= W_);
    if (interior) {
        const float* base = img + (y0 - RAD) * W_ + (x0 - RAD);
        #pragma unroll
        for (int j = 0; j < (PATCH * PATCH) / 32; ++j) {
            int i  = lane + 32 * j;
            int pr = i / PATCH, pc = i - pr * PATCH;   // tile-invariant, LICM-hoistable
            patch[pr * PSTRIDE + pc] = base[pr * W_ + pc];
        }
    } else {
        #pragma unroll
        for (int j = 0; j < (PATCH * PATCH) / 32; ++j) {
            int i  = lane + 32 * j;
            int pr = i / PATCH, pc = i - pr * PATCH;
            int gy = y0 + pr - RAD, gx = x0 + pc - RAD;
            bool inb = (gy >= 0) && (gy < H_) && (gx >= 0) && (gx < W_);
            int cy = min(max(gy, 0), H_ - 1);
            int cx = min(max(gx, 0), W_ - 1);
            float vv = img[cy * W_ + cx];
            patch[pr * PSTRIDE + pc] = inb ? vv : 0.0f;
        }
    }
    __syncthreads();

    v8f acc = {};
    const int arow = (lane & 15) * PSTRIDE + ((lane >> 4) << 1); // per-lane, loop-invariant A base
    #pragma unroll
    for (int c = 0; c < NCHUNK; ++c) {
        int dy = c / 6;
        int jc = (c - dy * 6) * 4;
        v2f a  = *(const v2f*)(&patch[arow + dy * PSTRIDE + jc]); // imm DS offset
        v2f bb = bpack[c * 32 + lane];                            // imm DS offset
        acc = __builtin_amdgcn_wmma_f32_16x16x4_f32(
                  false, a, false, bb, (short)0, acc, false, false);
    }
    __syncthreads();
    return acc;
}

// Pass 1: channel mean  m[i] = (x0+x1+x2)/3
__global__ void lcn_chan_mean(const float* __restrict__ x, float* __restrict__ m, int n)
{
    int i = blockIdx.x * blockDim.x + threadIdx.x;
    if (i >= n) return;
    const float* xp = x + (size_t)i * 3;
    m[i] = (xp[0] + xp[1] + xp[2]) * (1.0f / 3.0f);
}

// Pass 2: local_mean = conv(m) (WMMA), fused with s2 = mean_c((x - lm)^2)
__global__ void __launch_bounds__(32)
lcn_conv_mean(const float* __restrict__ m, const float* __restrict__ x,
              const float* __restrict__ w, float* __restrict__ lm,
              float* __restrict__ s2)
{
    __shared__ float patch[PATCH * PSTRIDE];
    __shared__ float wl2[KS * WPAD];
    __shared__ v2f   bpack[NCHUNK * 32];
    const int lane = threadIdx.x;
    build_bpack(w, bpack, wl2, lane);

    const int half = lane >> 4, n = lane & 15;
    for (int q = 0; q < TILES_PER_BLK; ++q) {
        int t  = blockIdx.x * TILES_PER_BLK + q;
        int b  = t / TILES_PER_IMG;
        int r  = t - b * TILES_PER_IMG;
        int y0 = (r / TILES_X) * TILE;
        int x0 = (r - (r / TILES_X) * TILES_X) * TILE;

        v8f acc = wmma_conv_tile(m + (size_t)b * (H_ * W_), bpack, patch, y0, x0, lane);

        // one 64-bit base per lane per tile; rr offsets fold into load/store imms
        size_t lbase = ((size_t)b * H_ + y0 + half * 8) * W_ + x0 + n;
        float* lmp       = lm + lbase;
        float* s2p       = s2 + lbase;
        const float* xp  = x  + lbase * 3;
        #pragma unroll
        for (int rr = 0; rr < 8; ++rr) {
            float lmv = acc[rr];
            lmp[rr * W_] = lmv;
            float v0  = xp[rr * (W_ * 3) + 0] - lmv;
            float v1  = xp[rr * (W_ * 3) + 1] - lmv;
            float v2v = xp[rr * (W_ * 3) + 2] - lmv;
            s2p[rr * W_] = (v0 * v0 + v1 * v1 + v2v * v2v) * (1.0f / 3.0f);
        }
    }
}

// Pass 3: sigma = sqrt(conv(s2)) (WMMA) + per-image sum(sigma)
__global__ void __launch_bounds__(32)
lcn_conv_var(const float* __restrict__ s2, const float* __restrict__ w,
             float* __restrict__ sigma, float* __restrict__ bsum)
{
    __shared__ float patch[PATCH * PSTRIDE];
    __shared__ float wl2[KS * WPAD];
    __shared__ v2f   bpack[NCHUNK * 32];
    const int lane = threadIdx.x;
    build_bpack(w, bpack, wl2, lane);

    const int half = lane >> 4, n = lane & 15;
    for (int q = 0; q < TILES_PER_BLK; ++q) {
        int t  = blockIdx.x * TILES_PER_BLK + q;
        int b  = t / TILES_PER_IMG;
        int r  = t - b * TILES_PER_IMG;
        int y0 = (r / TILES_X) * TILE;
        int x0 = (r - (r / TILES_X) * TILES_X) * TILE;

        v8f acc = wmma_conv_tile(s2 + (size_t)b * (H_ * W_), bpack, patch, y0, x0, lane);

        size_t lbase = ((size_t)b * H_ + y0 + half * 8) * W_ + x0 + n;
        float* sgp = sigma + lbase;
        float lsum = 0.0f;
        #pragma unroll
        for (int rr = 0; rr < 8; ++rr) {
            float sg = sqrtf(fmaxf(acc[rr], 0.0f));
            sgp[rr * W_] = sg;
            lsum += sg;
        }
        #pragma unroll
        for (int off = 16; off > 0; off >>= 1)
            lsum += __shfl_down(lsum, off, 32);
        if (lane == 0) atomicAdd(&bsum[b], lsum);
    }
}

// Pass 4: out = (x - lm) / max(mean_sigma, sigma)
__global__ void lcn_final(const float* __restrict__ x, const float* __restrict__ lm,
                          const float* __restrict__ sigma, const float* __restrict__ bsum,
                          float* __restrict__ out, int n)
{
    int i = blockIdx.x * blockDim.x + threadIdx.x;
    if (i >= n) return;
    int b = i >> 18;                                 // 512*512 = 2^18 pixels per image
    float ms  = bsum[b] * (1.0f / (H_ * W_));
    float den = fmaxf(ms, sigma[i]);
    float inv = 1.0f / den;
    float lmv = lm[i];
    const float* xp = x + (size_t)i * 3;
    float* op = out + (size_t)i * 3;
    op[0] = (xp[0] - lmv) * inv;
    op[1] = (xp[1] - lmv) * inv;
    op[2] = (xp[2] - lmv) * inv;
}

extern "C" void kernel_launch(void* const* d_in, const int* in_sizes, int n_in,
                              void* d_out, int out_size, void* d_ws, size_t ws_size,
                              hipStream_t stream) {
    const float* x = (const float*)d_in[0];   // [32,512,512,3] fp32
    const float* w = (const float*)d_in[1];   // [9,9,1,1] fp32
    float* out = (float*)d_out;               // [32,512,512,3] fp32

    float* ws    = (float*)d_ws;
    float* m_sig = ws;                         // m (pass1/2), reused as sigma (pass3/4)
    float* lm    = ws + (size_t)NPIX;          // local_mean
    float* s2b   = ws + 2 * (size_t)NPIX;      // variance field
    float* bsum  = ws + 3 * (size_t)NPIX;      // 32 per-image sigma sums

    hipMemsetAsync(bsum, 0, B_ * sizeof(float), stream);

    lcn_chan_mean<<<(NPIX + 255) / 256, 256, 0, stream>>>(x, m_sig, NPIX);
    lcn_conv_mean<<<NBLK, 32, 0, stream>>>(m_sig, x, w, lm, s2b);
    lcn_conv_var <<<NBLK, 32, 0, stream>>>(s2b, w, m_sig, bsum);
    lcn_final    <<<(NPIX + 255) / 256, 256, 0, stream>>>(x, lm, m_sig, bsum, out, NPIX);
}